// LayerCharCNN_38001870635636
// MI455X (gfx1250) — compile-verified
//
#include <hip/hip_runtime.h>

typedef __attribute__((ext_vector_type(2))) float v2f;
typedef __attribute__((ext_vector_type(8))) float v8f;

#define B_    32
#define S_    256
#define C_    50
#define L_    20
#define F_    30
#define W_    3
#define NROWS (B_ * S_)        // 8192 (b,s) rows
#define NTILE (NROWS / 16)     // 512 row-tiles of 16
#define P_    (L_ - W_ + 1)    // 18 conv positions
#define OUTC  (C_ * F_)        // 1500 output channels

__global__ __launch_bounds__(256) void charcnn_wmma_f32(
    const float* __restrict__ x,     // [NROWS, C_, L_]
    const float* __restrict__ wgt,   // [C_*F_, 1, W_]
    const float* __restrict__ bias,  // [C_*F_]
    float* __restrict__ out)         // [NROWS, C_*F_]
{
    const int lane   = threadIdx.x & 31;
    const int waveIB = threadIdx.x >> 5;
    const int wave   = blockIdx.x * 8 + waveIB;     // wave-uniform
    if (wave >= NTILE * C_) return;                 // uniform exit: EXEC stays full

    const int ntile = wave / C_;
    const int c     = wave - ntile * C_;

    const int  m   = lane & 15;       // row within tile / N column
    const bool hi  = lane >= 16;      // K-half select per WMMA A/B layout
    const int  row = ntile * 16 + m;

    // ---- Shifted per-lane row view ----
    // lo lanes (K=0/1): av[j] = X[m, j];  hi lanes (K=2/3): av[j] = X[m, j+2].
    // Then a0 = av[p], a1 = av[p+1] with *constant* indices -> no cndmask trees.
    // a1 on hi lanes is the padded K=3 slot: B's K=3 row is 0, so any finite
    // value is fine there.
    const float* xrow  = x + (size_t)row * (C_ * L_) + (size_t)c * L_;
    const float* xbase = xrow + (hi ? 2 : 0);       // stays 8B-aligned

    float av[19];
#pragma unroll
    for (int i = 0; i < 9; ++i) {                   // 9 x b64: elements 0..17
        float2 v = *(const float2*)(xbase + 2 * i); // hi: row elems 2..19 (in-bounds)
        av[2 * i + 0] = v.x;
        av[2 * i + 1] = v.y;
    }
    av[18] = xrow[18];  // unshifted for ALL lanes: lo needs X[m,18]; hi value is
                        // a finite don't-care (multiplied by B[K=3]=0). No OOB.

    // ---- B operands: B[k,f] = wgt[(c*F_+f)*W_ + k], K=3 row zeroed ----
    // ISA B 4x16 layout: VGPR0 = K0 (lanes 0-15) / K2 (lanes 16-31),
    //                    VGPR1 = K1 (lanes 0-15) / K3 (lanes 16-31).
    const float* wc = wgt + (size_t)c * F_ * W_;
    const int k0 = hi ? 2 : 0;

    v2f b0, b1;
    {
        const int f0 = m;                       // filter tile 0: f = 0..15
        b0[0] = wc[f0 * W_ + k0];
        b0[1] = hi ? 0.0f : wc[f0 * W_ + 1];    // K=3 row of B is zero

        const int  f1 = 16 + m;                 // filter tile 1: f = 16..31
        const bool v1 = (f1 < F_);
        const int  fc = v1 ? f1 : (F_ - 1);     // clamp to stay in-bounds
        float t0 = wc[fc * W_ + k0];
        float t1 = wc[fc * W_ + 1];
        b1[0] = v1 ? t0 : 0.0f;
        b1[1] = (v1 && !hi) ? t1 : 0.0f;
    }

    v8f czero = {};
    v8f mx0, mx1;

    // ---- p = 0: initialize the running max directly ----
    {
        v2f a; a[0] = av[0]; a[1] = av[1];
        mx0 = __builtin_amdgcn_wmma_f32_16x16x4_f32(
            false, a, false, b0, (short)0, czero, false, false);
        mx1 = __builtin_amdgcn_wmma_f32_16x16x4_f32(
            false, a, false, b1, (short)0, czero, false, false);
    }

    // ---- p = 1..17: WMMA + elementwise running max ----
#pragma unroll
    for (int p = 1; p < P_; ++p) {
        v2f a; a[0] = av[p]; a[1] = av[p + 1];
        v8f d0 = __builtin_amdgcn_wmma_f32_16x16x4_f32(
            false, a, false, b0, (short)0, czero, false, false);
        v8f d1 = __builtin_amdgcn_wmma_f32_16x16x4_f32(
            false, a, false, b1, (short)0, czero, false, false);
#pragma unroll
        for (int r = 0; r < 8; ++r) {
            mx0[r] = fmaxf(mx0[r], d0[r]);
            mx1[r] = fmaxf(mx1[r], d1[r]);
        }
    }

    // ---- Bias + store. C/D layout: VGPR r -> M = r + 8*hi, N = lane%16 ----
    const int   nCol  = m;
    const float bias0 = bias[c * F_ + nCol];
    const bool  v1    = (16 + nCol) < F_;
    const float bias1 = bias[c * F_ + (v1 ? (16 + nCol) : 0)];

    float* obase = out + (size_t)(ntile * 16) * OUTC + (size_t)c * F_;
#pragma unroll
    for (int r = 0; r < 8; ++r) {
        const int M = r + (hi ? 8 : 0);
        float* orow = obase + (size_t)M * OUTC;
        orow[nCol] = mx0[r] + bias0;
        if (v1) orow[16 + nCol] = mx1[r] + bias1;
    }
}

extern "C" void kernel_launch(void* const* d_in, const int* in_sizes, int n_in,
                              void* d_out, int out_size, void* d_ws, size_t ws_size,
                              hipStream_t stream) {
    (void)in_sizes; (void)n_in; (void)out_size; (void)d_ws; (void)ws_size;
    const float* x    = (const float*)d_in[0];   // [B,S,C,L] f32
    const float* wgt  = (const float*)d_in[1];   // [C*F,1,W] f32
    const float* bias = (const float*)d_in[2];   // [C*F]     f32
    float* out        = (float*)d_out;           // [B,S,C*F] f32

    const int totalWaves = NTILE * C_;           // 25600
    const int blocks     = (totalWaves + 7) / 8; // 3200 blocks of 8 waves (256 thr)
    charcnn_wmma_f32<<<blocks, 256, 0, stream>>>(x, wgt, bias, out);
}